// ModalEnseModel_9156870275317
// MI455X (gfx1250) — compile-verified
//
#include <hip/hip_runtime.h>

// ModalEnseModel fuse: out[b] = concat( [vis[b,:, :5], vis[b,:,5:]*s_b], lwir[b] )
// along the anchor dim. Pure streaming op: ~1.10 GB traffic, ~47 us floor at
// 23.3 TB/s. No matrix contraction exists in this op, so WMMA is not used;
// the MI455X-relevant paths here are wave32 b128 global load/store with
// non-temporal hints (streams are touched once and exceed the 192 MB L2)
// plus gfx1250 global_prefetch_b8 to deepen HBM pipelining.

typedef float v4f __attribute__((ext_vector_type(4)));

namespace {
constexpr int kB       = 32;
constexpr int kN       = 25200;
constexpr int kC       = 85;
constexpr int kPlane   = kN * kC;        // 2,142,000 floats per [N,C] plane
constexpr int kPlaneV4 = kPlane / 4;     // 535,500 float4 per plane (exact)
constexpr int kBatchV4 = 2 * kPlaneV4;   // float4 per output batch plane [2N,C]
constexpr int kTotalV4 = kB * kBatchV4;  // 34,272,000 float4 total
}  // namespace

__global__ __launch_bounds__(256) void modal_ense_fuse(
    const float* __restrict__ vis,
    const float* __restrict__ lwir,
    const float* __restrict__ aware,   // [B,1] per-image score
    float* __restrict__ out) {
  const int stride = (int)(gridDim.x * blockDim.x);
  for (int v = (int)(blockIdx.x * blockDim.x + threadIdx.x); v < kTotalV4;
       v += stride) {
    const int b = v / kBatchV4;          // magic-number division (constant)
    const int r = v - b * kBatchV4;      // float4 index within output batch plane

    v4f val;
    const v4f* src;
    if (r < kPlaneV4) {
      // ---- visible stream: scale class columns (5..84) by aware[b] ----
      src = reinterpret_cast<const v4f*>(vis) + (size_t)b * kPlaneV4 + r;
      // Prefetch one grid-stride iteration ahead (speculative; OOB is dropped).
      __builtin_prefetch(reinterpret_cast<const char*>(src) +
                             (size_t)stride * sizeof(v4f),
                         0, 0);
      val = __builtin_nontemporal_load(src);

      const float s  = aware[b];         // 32 distinct addrs -> L2-resident
      const int   e0 = r * 4;            // flat element index in plane
      const int   c0 = e0 % kC;          // column of lane-elem 0, in [0,84]
      // Column of element j is (c0+j) wrapped at 85; wrapped cols are <=2 (<5),
      // so "scale" is exactly: 5 <= c0+j < 85.
      v4f m;
      m.x = (c0     >= 5)               ? s : 1.0f;   // c0 <= 84 always
      m.y = (c0 + 1 >= 5 && c0 + 1 < kC) ? s : 1.0f;
      m.z = (c0 + 2 >= 5 && c0 + 2 < kC) ? s : 1.0f;
      m.w = (c0 + 3 >= 5 && c0 + 3 < kC) ? s : 1.0f;
      val *= m;
    } else {
      // ---- lwir stream: straight copy ----
      src = reinterpret_cast<const v4f*>(lwir) + (size_t)b * kPlaneV4 +
            (r - kPlaneV4);
      __builtin_prefetch(reinterpret_cast<const char*>(src) +
                             (size_t)stride * sizeof(v4f),
                         0, 0);
      val = __builtin_nontemporal_load(src);
    }

    v4f* dst = reinterpret_cast<v4f*>(out) + (size_t)b * kBatchV4 + r;
    __builtin_nontemporal_store(val, dst);
  }
}

extern "C" void kernel_launch(void* const* d_in, const int* in_sizes, int n_in,
                              void* d_out, int out_size, void* d_ws,
                              size_t ws_size, hipStream_t stream) {
  const float* vis   = (const float*)d_in[0];   // [B, N, C] f32
  const float* lwir  = (const float*)d_in[1];   // [B, N, C] f32
  const float* aware = (const float*)d_in[2];   // [B, 1]    f32
  float* out = (float*)d_out;                   // [B, 2N, C] f32

  // 256 threads = 8 wave32 per block; 16384 blocks -> ~8.2 float4 per thread
  // (grid-stride), enough in-flight waves to saturate 23.3 TB/s while keeping
  // the prefetch distance meaningful.
  dim3 block(256);
  dim3 grid(16384);
  modal_ense_fuse<<<grid, block, 0, stream>>>(vis, lwir, aware, out);
}